// LightDGC_37778532335722
// MI455X (gfx1250) — compile-verified
//
#include <hip/hip_runtime.h>
#include <math.h>

// ---------------------------------------------------------------------------
// LightDGC on MI455X (gfx1250, wave32):
//   out = relu(tanh(a*(M1 M2^T - M2 M1^T))) * topk32(cos-sim(mean_t X[0]))
// Output is 32-sparse per row, so the N x N "diff" GEMM is replaced by 262k
// length-256 dot products at the surviving top-k positions. The only dense
// GEMM is sim = Xn Xn^T (f32, V_WMMA_F32_16X16X4_F32) with top-k fused so the
// 268MB sim matrix never leaves registers/LDS. Xn (8MB) is L2-resident, so
// B tiles stream from L2; A slabs live in LDS with a conflict-free stride.
// Inner loop is software-pipelined (register double buffer) + next-tile
// prefetch (global_prefetch_b8) to hide L2 latency under the WMMA chain.
// ---------------------------------------------------------------------------

typedef float v2f __attribute__((ext_vector_type(2)));
typedef float v8f __attribute__((ext_vector_type(8)));

#define NN    8192
#define DD    256
#define TT    12
#define KK    32
#define ALPHAC 0.2f
#define NWAVE 4
#define ASTR  258   // padded LDS stride: lane r hits banks {2r, 2r+1} -> conflict-free b64

// ---------------- Kernel 1: X0 = mean_t X[0], row L2-normalize -------------
__global__ __launch_bounds__(256)
void k_mean_norm(const float* __restrict__ X, float* __restrict__ Xn) {
    const int n = blockIdx.x;
    const int d = threadIdx.x;            // 256 threads == DD
    float s = 0.0f;
#pragma unroll
    for (int t = 0; t < TT; ++t)
        s += X[((size_t)t * NN + n) * DD + d];   // batch 0 slice only
    s *= (1.0f / (float)TT);

    __shared__ float red[256];
    red[d] = s * s;
    __syncthreads();
    for (int off = 128; off > 0; off >>= 1) {
        if (d < off) red[d] += red[d + off];
        __syncthreads();
    }
    const float inv = 1.0f / fmaxf(sqrtf(red[0]), 1e-12f);
    Xn[(size_t)n * DD + d] = s * inv;
}

// ------- Kernel 2: sim = Xn Xn^T via WMMA f32 16x16x4, fused top-32 --------
__global__ __launch_bounds__(128)
void k_sim_topk(const float* __restrict__ Xn,
                float* __restrict__ tv_out, int* __restrict__ ti_out) {
    __shared__ float As[16 * ASTR];            // 16 A-rows of this block
    __shared__ float tileS[NWAVE][16][17];     // staged 16x16 sim tiles
    __shared__ float tv[NWAVE][16][KK];        // per-wave top-32 values (sorted desc)
    __shared__ int   ti[NWAVE][16][KK];        // per-wave top-32 indices

    const int tid  = threadIdx.x;
    const int wave = tid >> 5;
    const int lane = tid & 31;
    const int lr   = lane & 15;                // row-in-tile selector
    const int lh   = lane >> 4;                // half selector (K pair / M+8)
    const int i0   = blockIdx.x * 16;

    for (int idx = tid; idx < 16 * DD; idx += 128) {
        const int r = idx >> 8, c = idx & 255;
        As[r * ASTR + c] = Xn[(size_t)(i0 + r) * DD + c];
    }
    for (int idx = tid; idx < NWAVE * 16 * KK; idx += 128) {
        ((float*)tv)[idx] = -__builtin_inff();
        ((int*)ti)[idx]   = 0;
    }
    __syncthreads();

    const float* arow = &As[lr * ASTR];

    // every wave runs exactly (NN/16)/NWAVE = 128 iterations -> barriers legal
    for (int jt = wave; jt < NN / 16; jt += NWAVE) {
        const int j0 = jt * 16;
        const float* brow = &Xn[(size_t)(j0 + lr) * DD];   // L2-resident (8MB)

        // prefetch this wave's NEXT column tile (16KB = 128 lines, 4/lane)
        if (jt + NWAVE < NN / 16) {
            const float* nb = &Xn[(size_t)(j0 + NWAVE * 16 + lr) * DD];
#pragma unroll
            for (int pc = 0; pc < 4; ++pc)
                __builtin_prefetch(nb + lh * 128 + pc * 32, 0, 3);
        }

        v8f c = {};
        v2f aR[4], bR[4];
#pragma unroll
        for (int u = 0; u < 4; ++u) {          // prime chunk 0 (k = 0..15)
            const int k0 = 4 * u + 2 * lh;
            aR[u] = *(const v2f*)(arow + k0);
            bR[u] = *(const v2f*)(brow + k0);
        }
        for (int kc = 16; kc <= DD; kc += 16) {
            v2f aN[4], bN[4];
            if (kc < DD) {                     // issue next chunk's loads first
#pragma unroll
                for (int u = 0; u < 4; ++u) {
                    const int k0 = kc + 4 * u + 2 * lh;
                    aN[u] = *(const v2f*)(arow + k0);
                    bN[u] = *(const v2f*)(brow + k0);
                }
            }
#pragma unroll
            for (int u = 0; u < 4; ++u)        // 4-deep WMMA chain covers latency
                c = __builtin_amdgcn_wmma_f32_16x16x4_f32(
                        false, aR[u], false, bR[u], (short)0, c, false, false);
#pragma unroll
            for (int u = 0; u < 4; ++u) { aR[u] = aN[u]; bR[u] = bN[u]; }
        }

        // zero diagonal + stage tile (c[v] -> row v+8*lh, col lr)
#pragma unroll
        for (int v = 0; v < 8; ++v) {
            const int lrow = v + 8 * lh;
            const float val = ((i0 + lrow) == (j0 + lr)) ? 0.0f : c[v];
            tileS[wave][lrow][lr] = val;
        }
        __syncthreads();
        // lanes 0..15 of each wave own one row: sorted-insert top-32
        if (lane < 16) {
            float* tvr = tv[wave][lane];
            int*   tir = ti[wave][lane];
            for (int t = 0; t < 16; ++t) {
                const float val = tileS[wave][lane][t];
                if (val > tvr[KK - 1]) {
                    int p = KK - 1;
                    while (p > 0 && tvr[p - 1] < val) {
                        tvr[p] = tvr[p - 1]; tir[p] = tir[p - 1]; --p;
                    }
                    tvr[p] = val; tir[p] = j0 + t;
                }
            }
        }
        __syncthreads();
    }

    // merge the NWAVE sorted lists per row, emit final top-32
    if (tid < 16) {
        const int r = tid;
        float fv[KK]; int fi[KK];
#pragma unroll
        for (int q = 0; q < KK; ++q) { fv[q] = tv[0][r][q]; fi[q] = ti[0][r][q]; }
        for (int w = 1; w < NWAVE; ++w) {
            for (int t = 0; t < KK; ++t) {
                const float val = tv[w][r][t];
                if (!(val > fv[KK - 1])) break;   // source sorted: rest smaller
                int p = KK - 1;
                while (p > 0 && fv[p - 1] < val) {
                    fv[p] = fv[p - 1]; fi[p] = fi[p - 1]; --p;
                }
                fv[p] = val; fi[p] = ti[w][r][t];
            }
        }
#pragma unroll
        for (int q = 0; q < KK; ++q) {
            tv_out[(size_t)(i0 + r) * KK + q] = fv[q];
            ti_out[(size_t)(i0 + r) * KK + q] = fi[q];
        }
    }
}

// -------- Kernel 3: M = tanh(ALPHA * (E @ theta)) via WMMA f32 -------------
__global__ __launch_bounds__(32)
void k_adapt_m(const float* __restrict__ E, const float* __restrict__ theta,
               float* __restrict__ M) {
    const int lane = threadIdx.x;
    const int lr = lane & 15, lh = lane >> 4;
    const int i0 = blockIdx.x * 16;
    const int n0 = blockIdx.y * 16;
    const float* erow = &E[(size_t)(i0 + lr) * DD];
    v8f c = {};
#pragma unroll 4
    for (int kb = 0; kb < DD; kb += 4) {
        const int k0 = kb + 2 * lh;
        v2f a = *(const v2f*)(erow + k0);
        v2f b;
        b.x = theta[(size_t)k0 * DD + n0 + lr];
        b.y = theta[(size_t)(k0 + 1) * DD + n0 + lr];
        c = __builtin_amdgcn_wmma_f32_16x16x4_f32(
                false, a, false, b, (short)0, c, false, false);
    }
#pragma unroll
    for (int v = 0; v < 8; ++v) {
        const int gi = i0 + v + 8 * lh;
        M[(size_t)gi * DD + n0 + lr] = tanhf(ALPHAC * c[v]);
    }
}

// ---------------- Kernel 4: zero-fill the 268MB output ---------------------
__global__ __launch_bounds__(256)
void k_zero(float4* __restrict__ out, size_t n4) {
    const size_t i = (size_t)blockIdx.x * blockDim.x + threadIdx.x;
    if (i < n4) out[i] = make_float4(0.f, 0.f, 0.f, 0.f);
}

// --- Kernel 5: sparse scatter: out[i,j] = relu(tanh(a*diff_ij)) * val ------
__global__ __launch_bounds__(256)
void k_scatter(const float* __restrict__ M1, const float* __restrict__ M2,
               const float* __restrict__ tv, const int* __restrict__ ti,
               const float* __restrict__ aArr, float* __restrict__ out) {
    const int i = blockIdx.x;
    const int wave = threadIdx.x >> 5;
    const int lane = threadIdx.x & 31;
    const float a = aArr[0];

    // hoist row-i vectors: lane owns 8 contiguous floats (two b128 loads each)
    const float4 x1a = ((const float4*)&M1[(size_t)i * DD])[lane * 2];
    const float4 x1b = ((const float4*)&M1[(size_t)i * DD])[lane * 2 + 1];
    const float4 x2a = ((const float4*)&M2[(size_t)i * DD])[lane * 2];
    const float4 x2b = ((const float4*)&M2[(size_t)i * DD])[lane * 2 + 1];

    for (int k = wave; k < KK; k += 8) {
        const int   j   = ti[(size_t)i * KK + k];
        const float val = tv[(size_t)i * KK + k];
        const float4 y1a = ((const float4*)&M1[(size_t)j * DD])[lane * 2];
        const float4 y1b = ((const float4*)&M1[(size_t)j * DD])[lane * 2 + 1];
        const float4 y2a = ((const float4*)&M2[(size_t)j * DD])[lane * 2];
        const float4 y2b = ((const float4*)&M2[(size_t)j * DD])[lane * 2 + 1];
        float s = x1a.x * y2a.x + x1a.y * y2a.y + x1a.z * y2a.z + x1a.w * y2a.w
                + x1b.x * y2b.x + x1b.y * y2b.y + x1b.z * y2b.z + x1b.w * y2b.w
                - x2a.x * y1a.x - x2a.y * y1a.y - x2a.z * y1a.z - x2a.w * y1a.w
                - x2b.x * y1b.x - x2b.y * y1b.y - x2b.z * y1b.z - x2b.w * y1b.w;
        for (int off = 16; off > 0; off >>= 1)
            s += __shfl_xor(s, off, 32);
        if (lane == 0)
            out[(size_t)i * NN + j] = fmaxf(0.0f, tanhf(a * s)) * val;
    }
}

// ---------------------------------------------------------------------------
extern "C" void kernel_launch(void* const* d_in, const int* in_sizes, int n_in,
                              void* d_out, int out_size, void* d_ws, size_t ws_size,
                              hipStream_t stream) {
    const float* X   = (const float*)d_in[0];
    const float* E1  = (const float*)d_in[1];
    const float* E2  = (const float*)d_in[2];
    const float* th1 = (const float*)d_in[3];
    const float* th2 = (const float*)d_in[4];
    const float* a   = (const float*)d_in[5];
    float* out = (float*)d_out;

    float* ws  = (float*)d_ws;
    float* Xn  = ws;                          // N*D
    float* M1  = Xn + (size_t)NN * DD;        // N*D
    float* M2  = M1 + (size_t)NN * DD;        // N*D
    float* tv  = M2 + (size_t)NN * DD;        // N*K
    int*   ti  = (int*)(tv + (size_t)NN * KK);// N*K

    k_mean_norm<<<NN, 256, 0, stream>>>(X, Xn);
    k_sim_topk<<<NN / 16, 128, 0, stream>>>(Xn, tv, ti);

    dim3 gM(NN / 16, DD / 16);
    k_adapt_m<<<gM, 32, 0, stream>>>(E1, th1, M1);
    k_adapt_m<<<gM, 32, 0, stream>>>(E2, th2, M2);

    const size_t n4 = (size_t)NN * NN / 4;
    k_zero<<<(unsigned)(n4 / 256), 256, 0, stream>>>((float4*)out, n4);
    k_scatter<<<NN, 256, 0, stream>>>(M1, M2, tv, ti, a, out);
}